// FieldAwareNeuralFactorizationMachine_90769838833781
// MI455X (gfx1250) — compile-verified
//
#include <hip/hip_runtime.h>
#include <hip/hip_bf16.h>

// Problem constants (match reference)
constexpr int kB    = 16384;
constexpr int kN    = 24;
constexpr int kCARD = 10000;
constexpr int kD    = 16;
constexpr int kNI   = kN * (kN - 1) / 2;   // 276
constexpr int kH1   = 128;
constexpr int kH2   = 64;

typedef float v2f __attribute__((ext_vector_type(2)));
typedef float v8f __attribute__((ext_vector_type(8)));
typedef int   v4i __attribute__((vector_size(16)));   // matches builtin param type

#if __has_builtin(__builtin_amdgcn_global_load_async_to_lds_b128)
#define FFM_ASYNC_LDS 1
#endif

// ---------------------------------------------------------------------------
// Kernel 1: embedding gather + pairwise interactions + linear term.
// One workgroup (256 threads) per sample. Stage the 24 rows
// E[i, x[b,i], :, :] (1536 B each, contiguous) into LDS, preferring the
// CDNA5 async global->LDS path (ASYNCcnt) so copies bypass VGPRs entirely.
// This kernel is the HBM-bound part (~604 MB total gather traffic).
// ---------------------------------------------------------------------------
__global__ __launch_bounds__(256) void ffm_gather_inter(
    const int*   __restrict__ x,
    const float* __restrict__ E,
    const float* __restrict__ Wlin,
    const float* __restrict__ blin,
    float*       __restrict__ pairs,
    float*       __restrict__ lin)
{
    __shared__ float fac[kN * kN * kD];   // 24*24*16 f32 = 36 KB
    __shared__ int   xs[kN];
    __shared__ float linacc;

    const int b = blockIdx.x;
    const int t = threadIdx.x;

    if (t < kN) xs[t] = x[b * kN + t];
    if (t == 0) linacc = 0.0f;
    __syncthreads();

    // Copy 24 rows * 384 floats = 2304 x 16B, coalesced.
    constexpr int ROW4 = kN * kD / 4;      // 96 float4 per row
    constexpr int TOT4 = kN * ROW4;        // 2304
    for (int idx = t; idx < TOT4; idx += 256) {
        const int row = idx / ROW4;
        const int off = idx - row * ROW4;
        const float4* src = reinterpret_cast<const float4*>(
            E + ((size_t)row * kCARD + (size_t)xs[row]) * (size_t)(kN * kD)) + off;
        float4* dst = reinterpret_cast<float4*>(fac) + row * ROW4 + off;
#ifdef FFM_ASYNC_LDS
        __builtin_amdgcn_global_load_async_to_lds_b128(
            (v4i*)src,                                       // flat 64-bit source addr
            (__attribute__((address_space(3))) v4i*)dst,     // LDS destination
            0, 0);
#else
        *dst = *src;
#endif
    }
    if (t < kN) {
        atomicAdd(&linacc, Wlin[t * kCARD + xs[t]]);
    }
#ifdef FFM_ASYNC_LDS
#if __has_builtin(__builtin_amdgcn_s_wait_asynccnt)
    __builtin_amdgcn_s_wait_asynccnt(0);
#else
    asm volatile("s_wait_asynccnt 0x0" ::: "memory");
#endif
#endif
    __syncthreads();

    // 276 pair dot products over D=16 from LDS, read as b128.
    for (int p = t; p < kNI; p += 256) {
        int i = 0, rem = p, cnt = kN - 1;
        while (rem >= cnt) { rem -= cnt; ++i; --cnt; }
        const int j = i + 1 + rem;
        const float4* a4 = reinterpret_cast<const float4*>(&fac[(i * kN + j) * kD]);
        const float4* c4 = reinterpret_cast<const float4*>(&fac[(j * kN + i) * kD]);
        float s = 0.0f;
#pragma unroll
        for (int d4 = 0; d4 < kD / 4; ++d4) {
            const float4 av = a4[d4];
            const float4 cv = c4[d4];
            s += av.x * cv.x + av.y * cv.y + av.z * cv.z + av.w * cv.w;
        }
        pairs[(size_t)b * kNI + p] = s;
    }
    if (t == 0) lin[b] = linacc + blin[0];
}

// ---------------------------------------------------------------------------
// Kernel 2: MLP 276 -> 128 -> 64 -> 1 using V_WMMA_F32_16X16X4_F32.
// One wave (32 lanes) per 16-sample tile. Batch dim = WMMA M dim.
// Fragment layouts per CDNA5 ISA 32-bit conventions:
//   A (16x4):  lane half h, lane lo = row M; VGPR j holds K = j + 2h
//   B (4x16):  VGPR j, half h -> row K = j + 2h, col N = lo
//   C/D (16x16): VGPR r, half h -> row M = r + 8h, col N = lo
// Activations pass through padded LDS to re-layout D-frag -> A-frag.
// ---------------------------------------------------------------------------
__global__ __launch_bounds__(32) void ffm_mlp(
    const float* __restrict__ pairs,
    const float* __restrict__ lin,
    const float* __restrict__ W1, const float* __restrict__ b1,
    const float* __restrict__ W2, const float* __restrict__ b2,
    const float* __restrict__ W3, const float* __restrict__ b3,
    float*       __restrict__ out)
{
    __shared__ float h1s[16][kH1 + 1];   // pad to 129 floats: conflict-free A reads
    __shared__ float h2s[16][kH2 + 1];   // pad to 65

    const int lane = threadIdx.x;
    const int half = lane >> 4;
    const int lo   = lane & 15;
    const int s0   = blockIdx.x * 16;

    // ---------------- Layer 1: (16 x 276) @ (276 x 128) ----------------
    v8f acc[8];
#pragma unroll
    for (int nt = 0; nt < 8; ++nt) {
        const float bb = b1[nt * 16 + lo];        // bias along N, replicated over M
        acc[nt] = (v8f){bb, bb, bb, bb, bb, bb, bb, bb};
    }
    const float* prow = pairs + (size_t)(s0 + lo) * kNI;
    for (int k0 = 0; k0 < kNI; k0 += 4) {         // 276 = 69 * 4, exact
        const int ka = k0 + 2 * half;
        v2f a;
        a.x = prow[ka];
        a.y = prow[ka + 1];
#pragma unroll
        for (int nt = 0; nt < 8; ++nt) {
            v2f bf;
            bf.x = W1[ka * kH1 + nt * 16 + lo];
            bf.y = W1[(ka + 1) * kH1 + nt * 16 + lo];
            acc[nt] = __builtin_amdgcn_wmma_f32_16x16x4_f32(
                false, a, false, bf, (short)0, acc[nt], false, false);
        }
    }
    // ReLU + write D-fragments to LDS in [sample][unit] layout.
#pragma unroll
    for (int nt = 0; nt < 8; ++nt)
#pragma unroll
        for (int r = 0; r < 8; ++r)
            h1s[r + 8 * half][nt * 16 + lo] = fmaxf(acc[nt][r], 0.0f);
    __syncthreads();

    // ---------------- Layer 2: (16 x 128) @ (128 x 64) ----------------
    v8f acc2[4];
#pragma unroll
    for (int nt = 0; nt < 4; ++nt) {
        const float bb = b2[nt * 16 + lo];
        acc2[nt] = (v8f){bb, bb, bb, bb, bb, bb, bb, bb};
    }
    for (int k0 = 0; k0 < kH1; k0 += 4) {
        const int ka = k0 + 2 * half;
        v2f a;
        a.x = h1s[lo][ka];
        a.y = h1s[lo][ka + 1];
#pragma unroll
        for (int nt = 0; nt < 4; ++nt) {
            v2f bf;
            bf.x = W2[ka * kH2 + nt * 16 + lo];
            bf.y = W2[(ka + 1) * kH2 + nt * 16 + lo];
            acc2[nt] = __builtin_amdgcn_wmma_f32_16x16x4_f32(
                false, a, false, bf, (short)0, acc2[nt], false, false);
        }
    }
#pragma unroll
    for (int nt = 0; nt < 4; ++nt)
#pragma unroll
        for (int r = 0; r < 8; ++r)
            h2s[r + 8 * half][nt * 16 + lo] = fmaxf(acc2[nt][r], 0.0f);
    __syncthreads();

    // ---------------- Layer 3: (16 x 64) @ (64 x 1) + linear term -------
    if (lane < 16) {
        float s = b3[0] + lin[s0 + lane];
#pragma unroll
        for (int k = 0; k < kH2; ++k) s += h2s[lane][k] * W3[k];
        out[s0 + lane] = s;
    }
}

// ---------------------------------------------------------------------------
extern "C" void kernel_launch(void* const* d_in, const int* in_sizes, int n_in,
                              void* d_out, int out_size, void* d_ws, size_t ws_size,
                              hipStream_t stream) {
    const int*   x    = (const int*)  d_in[0];
    const float* E    = (const float*)d_in[1];
    const float* Wlin = (const float*)d_in[2];
    const float* blin = (const float*)d_in[3];
    const float* W1   = (const float*)d_in[4];
    const float* b1   = (const float*)d_in[5];
    const float* W2   = (const float*)d_in[6];
    const float* b2   = (const float*)d_in[7];
    const float* W3   = (const float*)d_in[8];
    const float* b3   = (const float*)d_in[9];
    float* out = (float*)d_out;

    float* pairs = (float*)d_ws;                      // kB * kNI floats (18.1 MB)
    float* lin   = pairs + (size_t)kB * kNI;          // kB floats

    ffm_gather_inter<<<kB, 256, 0, stream>>>(x, E, Wlin, blin, pairs, lin);
    ffm_mlp<<<kB / 16, 32, 0, stream>>>(pairs, lin, W1, b1, W2, b2, W3, b3, out);
}